// ChamferDistanceLoss_7301444403296
// MI455X (gfx1250) — compile-verified
//
#include <hip/hip_runtime.h>

// CDNA5 / gfx1250 Chamfer distance.
// dist^2 tile computed entirely inside chained V_WMMA_F32_16X16X4_F32:
//   A1 = (-2ax, -2ay, -2az, |a|^2), B1 = (bx, by, bz, 1)   -> -2 a.b + |a|^2
//   A2 = (1, 0, 0, 0),              B2 = (|b|^2, 0, 0, 0)  -> + |b|^2 (chained on C)
// Running min kept with raw v_min3_num_f32 (2 tiles per issue); the >=0 clamp
// commutes with min, so it is applied once after the tile loop.

typedef __attribute__((ext_vector_type(2))) float v2f;
typedef __attribute__((ext_vector_type(8))) float v8f;

#define NBATCH 8
#define NPTS   4096
#define NTILE  (NPTS / 16)        // 256 tiles of 16 points
#define WPB    4                  // waves per block (128 threads)

__global__ void chamfer_zero_kernel(float* acc) { acc[0] = 0.0f; }

__device__ __forceinline__ v8f dist_tile_wmma(v2f a1, v2f a2,
                                              const float* __restrict__ qc,
                                              bool half)
{
    const float bx = qc[0], by = qc[1], bz = qc[2];
    const float sq2 = bx * bx + by * by + bz * bz;
    v2f b1;
    b1.x = half ? bz   : bx;
    b1.y = half ? 1.0f : by;
    v2f b2;
    b2.x = half ? 0.0f : sq2;
    b2.y = 0.0f;

    v8f c = {};
    c = __builtin_amdgcn_wmma_f32_16x16x4_f32(
            false, a1, false, b1, (short)0, c, false, false);
    c = __builtin_amdgcn_wmma_f32_16x16x4_f32(
            false, a2, false, b2, (short)0, c, false, false);
    return c;
}

__global__ __launch_bounds__(32 * WPB)
void chamfer_dir_kernel(const float* __restrict__ P,
                        const float* __restrict__ Q,
                        float* __restrict__ acc)
{
    const int lane = threadIdx.x & 31;
    const int wave = threadIdx.x >> 5;
    const int gw   = blockIdx.x * WPB + wave;     // global row-tile id
    const int b    = gw / NTILE;
    const int mt   = gw % NTILE;

    const float* p1 = P + (size_t)b * NPTS * 3;
    const float* p2 = Q + (size_t)b * NPTS * 3;

    const bool half = (lane >> 4) != 0;  // half 0: K=0,1  half 1: K=2,3
    const int  l15  = lane & 15;

    // ---- A1: lane holds row M=l15; VGPR0/1 = K=2*half, 2*half+1
    const int m = mt * 16 + l15;
    const float* pr = p1 + m * 3;
    const float ax = pr[0], ay = pr[1], az = pr[2];
    const float sq1 = ax * ax + ay * ay + az * az;
    v2f a1;
    a1.x = half ? (-2.0f * az) : (-2.0f * ax);
    a1.y = half ? sq1          : (-2.0f * ay);
    // ---- A2: column of ones in K=0
    v2f a2;
    a2.x = half ? 0.0f : 1.0f;
    a2.y = 0.0f;

    float mn[8];
#pragma unroll
    for (int r = 0; r < 8; ++r) mn[r] = 3.4e38f;

    const float* qbase = p2 + l15 * 3;
    for (int nt = 0; nt < NTILE; nt += 2) {
        v8f c0 = dist_tile_wmma(a1, a2, qbase + (size_t)nt * 48, half);
        v8f c1 = dist_tile_wmma(a1, a2, qbase + (size_t)(nt + 1) * 48, half);
#pragma unroll
        for (int r = 0; r < 8; ++r) {
            float x = c0[r], y = c1[r];
            // mn = min3(mn, c0, c1) without fminf's sNaN-canonicalize maxes
            asm("v_min3_num_f32 %0, %0, %1, %2" : "+v"(mn[r]) : "v"(x), "v"(y));
        }
    }

    // Column (N) reduction: min across the 16 lanes of each half.
    // xor masks 1,2,4,8 never cross the half boundary (bit 4).
#pragma unroll
    for (int msk = 1; msk < 16; msk <<= 1) {
#pragma unroll
        for (int r = 0; r < 8; ++r)
            mn[r] = fminf(mn[r], __shfl_xor(mn[r], msk, 32));
    }

    float s = 0.0f;
#pragma unroll
    for (int r = 0; r < 8; ++r)
        s += sqrtf(fmaxf(mn[r], 0.0f));   // clamp fp slop once, then sqrt

    // lane 0 covers rows 0..7 of the tile, lane 16 covers rows 8..15
    if (l15 == 0) atomicAdd(acc, s);
}

__global__ void chamfer_final_kernel(const float* __restrict__ acc,
                                     float* __restrict__ out)
{
    // loss = mean_b[ mean_n dmin1 + mean_m dmin2 ] = (S1 + S2) / (B*N)
    out[0] = acc[0] * (1.0f / (float)(NBATCH * NPTS));
}

extern "C" void kernel_launch(void* const* d_in, const int* in_sizes, int n_in,
                              void* d_out, int out_size, void* d_ws, size_t ws_size,
                              hipStream_t stream)
{
    const float* p1 = (const float*)d_in[0];
    const float* p2 = (const float*)d_in[1];
    float* acc = (float*)d_ws;
    float* out = (float*)d_out;

    chamfer_zero_kernel<<<dim3(1), dim3(1), 0, stream>>>(acc);

    const int blocks = NBATCH * NTILE / WPB;   // 512
    chamfer_dir_kernel<<<dim3(blocks), dim3(32 * WPB), 0, stream>>>(p1, p2, acc);
    chamfer_dir_kernel<<<dim3(blocks), dim3(32 * WPB), 0, stream>>>(p2, p1, acc);

    chamfer_final_kernel<<<dim3(1), dim3(1), 0, stream>>>(acc, out);
}